// Routing_2259152797848
// MI455X (gfx1250) — compile-verified
//
#include <hip/hip_runtime.h>
#include <math.h>

#define CHANNEL 8
#define KDIM    16
#define DD      128        // CHANNEL * KDIM
#define M_NB    32
#define ROUT_IT 3
#define EPSN    1e-12f

// padded LDS strides (avoid 64-bank conflicts on stride-128 accesses)
#define WT_STRIDE 129      // Wt[k*129 + j]
#define AS_STRIDE 132      // As[row*132 + k]  (multiple of 4 -> float4 aligned)

typedef __attribute__((ext_vector_type(2))) float v2f;
typedef __attribute__((ext_vector_type(8))) float v8f;

__global__ void zero_row_kernel(float* __restrict__ z, int n) {
  int t = threadIdx.x;
  if (t < DD) z[(size_t)n * DD + t] = 0.0f;
}

// Phase 1: z[i,:] = per-capsule-normalize(relu(x @ W^T + b)), i < n.
// Block = 256 threads = 8 waves; block handles 64 rows; wave w owns column
// tile w (== capsule w). K-loop of 32x V_WMMA_F32_16X16X4_F32 per 16x16 tile.
__global__ void fc_norm_kernel(const float* __restrict__ x,
                               const float* __restrict__ W,
                               const float* __restrict__ bias,
                               float* __restrict__ z, int n) {
  extern __shared__ float smem[];
  float* Wt = smem;                       // [128][WT_STRIDE]: Wt[k][j] = W[j][k]
  float* As = smem + DD * WT_STRIDE;      // [64][AS_STRIDE]
  const int t    = threadIdx.x;
  const int lane = t & 31;
  const int wave = t >> 5;                // 0..7  (capsule / col tile)
  const int base = blockIdx.x * 64;

  // Stage W transposed (coalesced global read; one-time LDS scatter).
  for (int i = 0; i < 64; ++i) {
    int idx = i * 256 + t;                // = j*128 + k
    int j = idx >> 7, k = idx & 127;
    Wt[k * WT_STRIDE + j] = W[idx];
  }
  // Stage 64-row A tile, zero-padded past n.
  for (int i = 0; i < 8; ++i) {
    int f4  = i * 256 + t;                // float4 id, < 2048
    int row = f4 >> 5;
    int kk  = (f4 & 31) << 2;
    float4 v = make_float4(0.f, 0.f, 0.f, 0.f);
    int rg = base + row;
    if (rg < n) v = *(const float4*)(x + (size_t)rg * DD + kk);
    *(float4*)(As + row * AS_STRIDE + kk) = v;
  }
  __syncthreads();

  const int   col0 = wave * KDIM;
  const int   half = lane >> 4;           // A: K+= {0,2}; C: M += {0,8}
  const int   l16  = lane & 15;
  const float bcol = bias[col0 + l16];

  for (int rt = 0; rt < 4; ++rt) {
    const int row0 = rt * 16;
    v8f acc = {};
    for (int k0 = 0; k0 < DD; k0 += 4) {
      const int ka = k0 + half * 2;
      v2f a, bb;
      // A 16x4 f32: lanes 0-15 hold K=k0,k0+1 of row l16; lanes 16-31 K=k0+2,k0+3
      a[0] = As[(row0 + l16) * AS_STRIDE + ka];
      a[1] = As[(row0 + l16) * AS_STRIDE + ka + 1];
      // B 4x16 f32: B[k][j] = W[j][k] = Wt[k][j]
      bb[0] = Wt[ka * WT_STRIDE + col0 + l16];
      bb[1] = Wt[(ka + 1) * WT_STRIDE + col0 + l16];
      acc = __builtin_amdgcn_wmma_f32_16x16x4_f32(false, a, false, bb,
                                                  (short)0, acc, false, false);
    }
    // Epilogue: bias + relu + L2-normalize over the 16 cols (== capsule `wave`).
    // C layout: VGPR v -> row (row0 + v + 8*half), lane l16 -> column.
#pragma unroll
    for (int v = 0; v < 8; ++v) {
      float val = fmaxf(acc[v] + bcol, 0.0f);
      float ss = val * val;                       // reduce over 16-lane half
      ss += __shfl_xor(ss, 1, 32);
      ss += __shfl_xor(ss, 2, 32);
      ss += __shfl_xor(ss, 4, 32);
      ss += __shfl_xor(ss, 8, 32);
      float inv = 1.0f / fmaxf(sqrtf(ss), EPSN);  // v / max(||v||, eps)
      int rg = base + row0 + v + half * 8;
      if (rg < n) z[(size_t)rg * DD + col0 + l16] = val * inv;
    }
  }
}

// Phase 2: dynamic routing. One wave per node; 32 neighbor rows (16 KB) staged
// in a per-wave LDS region once, reused for all 3 iterations. Lane l holds
// float4 of u for channel c = l/4, k-quarter q = l%4. Softmax over channels is
// fused per neighbor (exact: softmax axis is channels, per (node,m)).
__global__ void routing_kernel(const float* __restrict__ z,
                               const int* __restrict__ nbid,
                               float* __restrict__ out, int n) {
  extern __shared__ float nbuf[];               // 8 waves * 4096 floats
  const int lane = threadIdx.x & 31;
  const int wave = threadIdx.x >> 5;
  const int node = blockIdx.x * 8 + wave;
  if (node >= n) return;
  float* my = nbuf + wave * (M_NB * DD);

  // Fill: one full 512B neighbor row per m (wave-wide float4), L2-resident z.
  const int nb_l = nbid[(size_t)node * M_NB + lane];
  for (int m = 0; m < M_NB; ++m) {
    int id = __shfl(nb_l, m, 32);               // id==n hits the zero pad row
    float4 v = *(const float4*)(z + (size_t)id * DD + lane * 4);
    *(float4*)(my + m * DD + lane * 4) = v;
  }
  __builtin_amdgcn_wave_barrier();
  asm volatile("s_wait_dscnt 0x0" ::: "memory"); // per-wave LDS RAW fence

  const int off = (lane >> 2) * KDIM + (lane & 3) * 4;  // c*16 + q*4
  float4 xc = *(const float4*)(z + (size_t)node * DD + off);
  float4 u = xc;

  for (int it = 0; it < ROUT_IT; ++it) {
    float4 acc = make_float4(0.f, 0.f, 0.f, 0.f);
    for (int m = 0; m < M_NB; ++m) {
      float4 nv = *(const float4*)(my + m * DD + off);
      // p[m][c] = <u_c, neigh_c>: reduce dot4 over the 4 lanes of channel c
      float pd = u.x * nv.x + u.y * nv.y + u.z * nv.z + u.w * nv.w;
      pd += __shfl_xor(pd, 1, 32);
      pd += __shfl_xor(pd, 2, 32);
      // softmax over the 8 channels (xor 4/8/16 hops across channel groups)
      float mx = pd;
      mx = fmaxf(mx, __shfl_xor(mx, 4, 32));
      mx = fmaxf(mx, __shfl_xor(mx, 8, 32));
      mx = fmaxf(mx, __shfl_xor(mx, 16, 32));
      float e = __expf(pd - mx);                 // TAU == 1
      float s = e;
      s += __shfl_xor(s, 4, 32);
      s += __shfl_xor(s, 8, 32);
      s += __shfl_xor(s, 16, 32);
      float p = e / s;
      acc.x += p * nv.x; acc.y += p * nv.y;
      acc.z += p * nv.z; acc.w += p * nv.w;
    }
    u.x = acc.x + xc.x; u.y = acc.y + xc.y;
    u.z = acc.z + xc.z; u.w = acc.w + xc.w;
    if (it < ROUT_IT - 1) {                      // normalize per capsule
      float ss = u.x * u.x + u.y * u.y + u.z * u.z + u.w * u.w;
      ss += __shfl_xor(ss, 1, 32);
      ss += __shfl_xor(ss, 2, 32);
      float inv = 1.0f / fmaxf(sqrtf(ss), EPSN);
      u.x *= inv; u.y *= inv; u.z *= inv; u.w *= inv;
    }
  }
  *(float4*)(out + (size_t)node * DD + off) = u;
}

extern "C" void kernel_launch(void* const* d_in, const int* in_sizes, int n_in,
                              void* d_out, int out_size, void* d_ws, size_t ws_size,
                              hipStream_t stream) {
  const float* x    = (const float*)d_in[0];
  const float* W    = (const float*)d_in[1];
  const float* bias = (const float*)d_in[2];
  const int*   nbid = (const int*)d_in[3];
  float*       out  = (float*)d_out;
  const int n = in_sizes[0] / DD;

  float* z = (float*)d_ws;                       // (n+1) x 128 normalized acts

  zero_row_kernel<<<1, 128, 0, stream>>>(z, n);

  const size_t lds1 = (size_t)(DD * WT_STRIDE + 64 * AS_STRIDE) * sizeof(float);
  fc_norm_kernel<<<(n + 63) / 64, 256, lds1, stream>>>(x, W, bias, z, n);

  const size_t lds2 = (size_t)8 * M_NB * DD * sizeof(float);  // 128 KB
  routing_kernel<<<(n + 7) / 8, 256, lds2, stream>>>(z, nbid, out, n);
}